// MaxwellBornTM_19516331393776
// MI455X (gfx1250) — compile-verified
//
#include <hip/hip_runtime.h>

// ---------------- problem constants (match reference) ----------------
#define NYD 288
#define NXD 288
#define NSH 2            // shots
#define NSRCD 1
#define NRECD 64
#define NTD 96
#define NCD (NYD * NXD)
#define NSCD (NSH * NCD)
#define PMLW 20

#define EP0f  8.8541878128e-12f
#define MU0Cf 1.25663706212e-6f
#define DTf   4.0e-11f
#define RDf   100.0f      // 1/DX = 1/DY

// tile geometry: 32x32 owned cells, k=4 steps per launch -> 4-cell halo
#define TLE 32
#define KSTEP 4
#define HAL KSTEP
#define FR4 (TLE + 2 * HAL)    // 40 staged frame
#define FS4 (FR4 + 1)          // 41 padded row stride (odd -> bank-conflict-free)
#define FSZ (FR4 * FS4)        // 1640 floats per staged array

__device__ __forceinline__ int clampi(int v, int lo, int hi) {
  return v < lo ? lo : (v > hi ? hi : v);
}

// Low 32 bits of a generic pointer to LDS == byte offset within the wave's
// LDS allocation (ISA 10.2 aperture rules) -> usable as async LDS operand.
__device__ __forceinline__ unsigned lds_off(const void* p) {
  return (unsigned)(unsigned long long)p;
}

// CDNA5 async global<->LDS copies (ASYNCcnt-tracked), per-lane addresses.
__device__ __forceinline__ void async_ld_f32(unsigned lds, const float* g) {
  asm volatile("global_load_async_to_lds_b32 %0, %1, off"
               :: "v"(lds), "v"(g) : "memory");
}
__device__ __forceinline__ void async_st_f32(float* g, unsigned lds) {
  asm volatile("global_store_async_from_lds_b32 %0, %1, off"
               :: "v"(g), "v"(lds) : "memory");
}
__device__ __forceinline__ void wait_async0() {
  asm volatile("s_wait_asynccnt 0x0" ::: "memory");
}

// ---------------------------------------------------------------------
// Workspace layout (floats):
//   bufA : 14 * NSCD   (Eb,Es,Hxb,Hzb,Hxs,Hzs,mHxb,mHzb,mHxs,mHzs,mExb,mEzb,mExs,mEzs)
//   bufB : 14 * NSCD   (ping-pong buffer, swapped every KSTEP steps)
//   coef :  6 * NCD    (ca0, cb0, cq0, dca, dcb, dcq)
//   prof :  4 * NYD    (a, ah, b, bh)  [NY==NX, DY==DX -> shared profiles]
// ---------------------------------------------------------------------

__global__ void born_init(const float* __restrict__ eps_r, const float* __restrict__ sig0,
                          const float* __restrict__ mu_r, const float* __restrict__ deps_r,
                          const float* __restrict__ dsig, const float* __restrict__ dmu_r,
                          float* __restrict__ bufA, float* __restrict__ bufB,
                          float* __restrict__ coef, float* __restrict__ prof) {
  int i = blockIdx.x * blockDim.x + threadIdx.x;
  if (i < NSCD) {
#pragma unroll
    for (int f = 0; f < 14; ++f) {
      bufA[(size_t)f * NSCD + i] = 0.0f;
      bufB[(size_t)f * NSCD + i] = 0.0f;
    }
  }
  if (i < NCD) {
    float eps = eps_r[i] * EP0f;
    float mu  = mu_r[i] * MU0Cf;
    float half = sig0[i] * DTf / (2.0f * eps);
    float den = 1.0f + half;
    float cb0 = (DTf / eps) / den;
    coef[0 * NCD + i] = (1.0f - half) / den;                 // ca0
    coef[1 * NCD + i] = cb0;                                 // cb0
    coef[2 * NCD + i] = DTf / mu;                            // cq0
    float de = deps_r[i] * EP0f;
    float dm = dmu_r[i] * MU0Cf;
    float d2 = den * den;
    coef[3 * NCD + i] = (sig0[i] * DTf / (eps * eps * d2)) * de
                      + (-DTf / (eps * d2)) * dsig[i];       // dca
    float t1 = 1.0f / den, t2 = half / d2;
    coef[4 * NCD + i] = (-(DTf / (eps * eps)) * (t1 + t2)) * de
                      + (-(DTf / eps) * (DTf / (2.0f * eps)) / d2) * dsig[i]; // dcb
    coef[5 * NCD + i] = (-DTf / (mu * mu)) * dm;             // dcq
  }
  if (i < NYD) {
    // smax = -(STENCIL+1)*MAX_VEL*ln(1e-5)/(2*PML*DX)
    const float SMAX = (float)(3.0 * 1.5e8 * 11.512925464970229 / (2.0 * PMLW * 0.01));
    float p0 = (float)i;
    float d0 = fmaxf((float)PMLW - p0, p0 - (float)(NYD - 1 - PMLW)) * (1.0f / (float)PMLW);
    d0 = fminf(fmaxf(d0, 0.0f), 1.0f);
    float s0 = SMAX * d0 * d0;
    float b0 = expf(-s0 * DTf);
    prof[0 * NYD + i] = (s0 > 0.0f) ? (b0 - 1.0f) : 0.0f;    // a
    prof[2 * NYD + i] = b0;                                  // b
    float ph = p0 + 0.5f;
    float dh = fmaxf((float)PMLW - ph, ph - (float)(NYD - 1 - PMLW)) * (1.0f / (float)PMLW);
    dh = fminf(fmaxf(dh, 0.0f), 1.0f);
    float sh = SMAX * dh * dh;
    float bh = expf(-sh * DTf);
    prof[1 * NYD + i] = (sh > 0.0f) ? (bh - 1.0f) : 0.0f;    // ah
    prof[3 * NYD + i] = bh;                                  // bh
  }
}

// KSTEP fused time steps per launch. All 14 fields + 6 coefficient fields
// staged (async) into LDS over a 40x40 frame; H/E updates iterate in LDS on
// a region shrinking 1 cell/side per full step; receivers sampled from LDS
// each sub-step; owned 32x32 written back via async LDS->global stores.
__global__ void __launch_bounds__(256)
born_step4(const float* __restrict__ fin, float* __restrict__ fout,
           const float* __restrict__ coef, const float* __restrict__ prof,
           const float* __restrict__ amp, const int* __restrict__ src_idx,
           const int* __restrict__ rec_idx, float* __restrict__ out, int t0) {
  const int s   = blockIdx.z;
  const int ty0 = blockIdx.y * TLE;
  const int tx0 = blockIdx.x * TLE;
  const int tid = threadIdx.x;

  const float* pa  = prof + 0 * NYD; const float* pah = prof + 1 * NYD;
  const float* pb  = prof + 2 * NYD; const float* pbh = prof + 3 * NYD;

  __shared__ float smem[20 * FSZ];
  float* sEb  = smem +  0 * FSZ; float* sEs  = smem +  1 * FSZ;
  float* sHxb = smem +  2 * FSZ; float* sHzb = smem +  3 * FSZ;
  float* sHxs = smem +  4 * FSZ; float* sHzs = smem +  5 * FSZ;
  float* sMHxb = smem + 6 * FSZ; float* sMHzb = smem + 7 * FSZ;
  float* sMHxs = smem + 8 * FSZ; float* sMHzs = smem + 9 * FSZ;
  float* sMExb = smem + 10 * FSZ; float* sMEzb = smem + 11 * FSZ;
  float* sMExs = smem + 12 * FSZ; float* sMEzs = smem + 13 * FSZ;
  float* sCA  = smem + 14 * FSZ; float* sCB  = smem + 15 * FSZ;
  float* sCQ  = smem + 16 * FSZ; float* sDCA = smem + 17 * FSZ;
  float* sDCB = smem + 18 * FSZ; float* sDCQ = smem + 19 * FSZ;

  // ---- stage all fields + coefficients (clamped halo) via async copies ----
  for (int i = tid; i < FR4 * FR4; i += 256) {
    int fy = i / FR4, fx = i % FR4;          // frame <-> cell (ty0+fy-HAL, tx0+fx-HAL)
    int gy = clampi(ty0 + fy - HAL, 0, NYD - 1);
    int gx = clampi(tx0 + fx - HAL, 0, NXD - 1);
    int gg = gy * NXD + gx;
    int g  = s * NCD + gg;
    unsigned l0 = lds_off(&smem[fy * FS4 + fx]);
#pragma unroll
    for (int f = 0; f < 14; ++f)
      async_ld_f32(l0 + (unsigned)(f * FSZ * 4), fin + (size_t)f * NSCD + g);
#pragma unroll
    for (int c = 0; c < 6; ++c)
      async_ld_f32(l0 + (unsigned)((14 + c) * FSZ * 4), coef + (size_t)c * NCD + gg);
  }
  wait_async0();
  __syncthreads();

  for (int j = 1; j <= KSTEP; ++j) {
    const int tt = t0 + j - 1;
    // ---- H half-update on frame region [j-1 .. 39-j]^2 (in place) ----
    {
      const int lo = j - 1, w = FR4 + 1 - 2 * j;   // 41-2j
      for (int i = tid; i < w * w; i += 256) {
        int fy = lo + i / w, fx = lo + i % w;
        int fidx = fy * FS4 + fx;
        int gy = ty0 + fy - HAL, gx = tx0 + fx - HAL;
        int cgy = clampi(gy, 0, NYD - 1), cgx = clampi(gx, 0, NXD - 1);
        float cq = sCQ[fidx], dq = sDCQ[fidx];
        float bhy = pbh[cgy], ahy = pah[cgy];
        float bhx = pbh[cgx], ahx = pah[cgx];
        // dfy(E): zero-padded at y == NY-1 (reference jnp.pad semantics)
        float gb = (gy >= NYD - 1) ? 0.0f : (sEb[fidx + FS4] - sEb[fidx]) * RDf;
        float gs = (gy >= NYD - 1) ? 0.0f : (sEs[fidx + FS4] - sEs[fidx]) * RDf;
        float mxb = bhy * sMHxb[fidx] + ahy * gb;  float e_b = gb + mxb;
        float mxs = bhy * sMHxs[fidx] + ahy * gs;  float e_s = gs + mxs;
        float hxb = sHxb[fidx] - cq * e_b;
        float hxs = sHxs[fidx] - cq * e_s - dq * e_b;
        // dfx(E): zero-padded at x == NX-1
        float fbv = (gx >= NXD - 1) ? 0.0f : (sEb[fidx + 1] - sEb[fidx]) * RDf;
        float fsv = (gx >= NXD - 1) ? 0.0f : (sEs[fidx + 1] - sEs[fidx]) * RDf;
        float mzb = bhx * sMHzb[fidx] + ahx * fbv; float f_b = fbv + mzb;
        float mzs = bhx * sMHzs[fidx] + ahx * fsv; float f_s = fsv + mzs;
        float hzb = sHzb[fidx] + cq * f_b;
        float hzs = sHzs[fidx] + cq * f_s + dq * f_b;
        sMHxb[fidx] = mxb; sMHxs[fidx] = mxs; sMHzb[fidx] = mzb; sMHzs[fidx] = mzs;
        sHxb[fidx] = hxb; sHxs[fidx] = hxs; sHzb[fidx] = hzb; sHzs[fidx] = hzs;
      }
    }
    __syncthreads();
    // ---- E update on frame region [j .. 39-j]^2 (in place) ----
    {
      const int lo = j, w = FR4 - 2 * j;           // 40-2j
      const int sc = src_idx[s];                   // NSRC == 1
      const float av = amp[s * NTD + tt];
      for (int i = tid; i < w * w; i += 256) {
        int fy = lo + i / w, fx = lo + i % w;
        int fidx = fy * FS4 + fx;
        int gy = ty0 + fy - HAL, gx = tx0 + fx - HAL;
        int cgy = clampi(gy, 0, NYD - 1), cgx = clampi(gx, 0, NXD - 1);
        float pax = pa[cgx], pbx = pb[cgx];
        float pay = pa[cgy], pby = pb[cgy];
        // dbx_(Hz): zero-padded at x == 0 ; dby_(Hx): zero-padded at y == 0
        float cxbg = (gx == 0) ? 0.0f : (sHzb[fidx] - sHzb[fidx - 1]) * RDf;
        float mxb = pbx * sMExb[fidx] + pax * cxbg; float cxb = cxbg + mxb;
        float czbg = (gy == 0) ? 0.0f : (sHxb[fidx] - sHxb[fidx - FS4]) * RDf;
        float mzb = pby * sMEzb[fidx] + pay * czbg; float czb = czbg + mzb;
        float curlb = cxb - czb;
        float cxsg = (gx == 0) ? 0.0f : (sHzs[fidx] - sHzs[fidx - 1]) * RDf;
        float mxs = pbx * sMExs[fidx] + pax * cxsg; float cxs = cxsg + mxs;
        float czsg = (gy == 0) ? 0.0f : (sHxs[fidx] - sHxs[fidx - FS4]) * RDf;
        float mzs = pby * sMEzs[fidx] + pay * czsg; float czs = czsg + mzs;
        float curls = cxs - czs;
        float ebo = sEb[fidx], eso = sEs[fidx];
        float ca = sCA[fidx], cb = sCB[fidx];
        float es_n = ca * eso + cb * curls + sDCA[fidx] * ebo + sDCB[fidx] * curlb;
        float eb_n = ca * ebo + cb * curlb;
        // soft source wherever this cell is computed (incl. halo recompute)
        if (gy >= 0 && gy < NYD && gx >= 0 && gx < NXD && sc == gy * NXD + gx)
          eb_n += av * cb;
        sMExb[fidx] = mxb; sMEzb[fidx] = mzb; sMExs[fidx] = mxs; sMEzs[fidx] = mzs;
        sEb[fidx] = eb_n; sEs[fidx] = es_n;
      }
    }
    __syncthreads();
    // ---- receivers (owned region only; reads E arrays, next H writes H/mH) ----
    for (int rI = tid; rI < NRECD; rI += 256) {
      int cell = rec_idx[s * NRECD + rI];
      int ry = cell / NXD, rx = cell - (cell / NXD) * NXD;
      if (ry >= ty0 && ry < ty0 + TLE && rx >= tx0 && rx < tx0 + TLE) {
        int fidx = (ry - ty0 + HAL) * FS4 + (rx - tx0 + HAL);
        out[((0 * NSH + s) * NRECD + rI) * NTD + tt] = sEb[fidx];
        out[((1 * NSH + s) * NRECD + rI) * NTD + tt] = sEs[fidx];
      }
    }
  }

  // ---- write back owned 32x32 of all 14 fields via async LDS->global ----
  for (int i = tid; i < TLE * TLE; i += 256) {
    int oy = i / TLE, ox = i % TLE;
    int fidx = (oy + HAL) * FS4 + (ox + HAL);
    int idx = s * NCD + (ty0 + oy) * NXD + (tx0 + ox);
    unsigned l0 = lds_off(&smem[fidx]);
#pragma unroll
    for (int f = 0; f < 14; ++f)
      async_st_f32(fout + (size_t)f * NSCD + idx, l0 + (unsigned)(f * FSZ * 4));
  }
  wait_async0();   // stores complete (S_ENDPGM also implies wait-idle)
}

extern "C" void kernel_launch(void* const* d_in, const int* in_sizes, int n_in,
                              void* d_out, int out_size, void* d_ws, size_t ws_size,
                              hipStream_t stream) {
  (void)in_sizes; (void)n_in; (void)out_size; (void)ws_size;
  const float* eps0 = (const float*)d_in[0];
  const float* sig0 = (const float*)d_in[1];
  const float* mu0  = (const float*)d_in[2];
  const float* deps = (const float*)d_in[3];
  const float* dsig = (const float*)d_in[4];
  const float* dmu  = (const float*)d_in[5];
  const float* amp  = (const float*)d_in[6];
  const int* src_idx = (const int*)d_in[7];
  const int* rec_idx = (const int*)d_in[8];
  float* out = (float*)d_out;

  float* ws   = (float*)d_ws;
  float* bufA = ws;
  float* bufB = ws + (size_t)14 * NSCD;
  float* coef = ws + (size_t)28 * NSCD;
  float* prof = coef + (size_t)6 * NCD;

  born_init<<<(NSCD + 255) / 256, 256, 0, stream>>>(eps0, sig0, mu0, deps, dsig, dmu,
                                                    bufA, bufB, coef, prof);

  dim3 grid(NXD / TLE, NYD / TLE, NSH);
  for (int t0 = 0, p = 0; t0 < NTD; t0 += KSTEP, ++p) {
    const float* fin = (p & 1) ? bufB : bufA;
    float* fout = (p & 1) ? bufA : bufB;
    born_step4<<<grid, 256, 0, stream>>>(fin, fout, coef, prof, amp,
                                         src_idx, rec_idx, out, t0);
  }
}